// LyapunovSDELayer_33028298506966
// MI455X (gfx1250) — compile-verified
//
#include <hip/hip_runtime.h>
#include <hip/hip_bf16.h>

typedef __attribute__((ext_vector_type(2))) float v2f;
typedef __attribute__((ext_vector_type(8))) float v8f;

#define THETA 0.3f
// decay per step = 1 - KAPPA = 0.5  ->  path[m,t] = THETA + 0.5^t * (lam0[m]-THETA)

// One wave = 16 rows x full 256-step horizon, produced as 16 WMMA rank-1 tiles.
// Block = 256 threads = 8 waves = 128 rows.
__global__ __launch_bounds__(256) void LyapunovSDELayer_kernel(
    const float* __restrict__ lam, float* __restrict__ out) {
  const int tid  = threadIdx.x;
  const int lane = tid & 31;
  const int wave = tid >> 5;
  const int m    = lane & 15;        // row-in-tile / col-in-tile index
  const bool lo  = (lane < 16);
  const int rowBase = blockIdx.x * 128 + wave * 16;

  // Every lane loads its tile's lam0 (lanes 16-31 re-load same 16 values: broadcast hit).
  const float lam0 = lam[rowBase + m];

  // A (16x4 f32): only K=0 column non-zero -> VGPR0, lanes 0-15.
  v2f a;
  a.x = lo ? (lam0 - THETA) : 0.0f;
  a.y = 0.0f;

  // B (4x16 f32): only K=0 row non-zero -> VGPR0, lanes 0-15 hold 0.5^n, n = lane.
  float bx = lo ? __builtin_bit_cast(float, (unsigned)(127 - m) << 23) : 0.0f;

  // C = THETA broadcast.
  v8f c;
#pragma unroll
  for (int i = 0; i < 8; ++i) c[i] = THETA;

  // D-VGPR v maps to row (v + 8*(lane>=16)), column m of the tile.
  float* p = out + (size_t)(rowBase + (lo ? 0 : 8)) * 256 + m;

#pragma unroll
  for (int j = 0; j < 16; ++j) {           // t-tiles: t0 = 16*j
    v2f b;
    b.x = bx;
    b.y = 0.0f;
    // D = A x B + C  ==  THETA + 0.5^(t0+n) * (lam0[m]-THETA)
    v8f d = __builtin_amdgcn_wmma_f32_16x16x4_f32(
        /*neg_a=*/false, a, /*neg_b=*/false, b,
        /*c_mod=*/(short)0, c, /*reuse_a=*/false, /*reuse_b=*/false);
    const int t0 = j * 16;
#pragma unroll
    for (int v = 0; v < 8; ++v) p[v * 256 + t0] = d[v];
    bx *= 0x1p-16f;                        // advance powers by 16 steps (exact)
  }

  // Column t=0 must equal lam0 bit-exactly (reference emits lam0 unmodified).
  if (lo) out[(size_t)(rowBase + m) * 256] = lam0;
}

extern "C" void kernel_launch(void* const* d_in, const int* in_sizes, int n_in,
                              void* d_out, int out_size, void* d_ws, size_t ws_size,
                              hipStream_t stream) {
  const float* lam = (const float*)d_in[0];   // (131072, 1) f32
  float* out = (float*)d_out;                 // (131072, 256) f32 row-major
  const int N = in_sizes[0];                  // 131072
  const int rowsPerBlock = 128;               // 8 waves * 16 rows
  const int grid = N / rowsPerBlock;
  LyapunovSDELayer_kernel<<<grid, 256, 0, stream>>>(lam, out);
}